// PureGCN_no_para_34720515620918
// MI455X (gfx1250) — compile-verified
//
#include <hip/hip_runtime.h>

// ---------------------------------------------------------------------------
// PureGCN (2-layer, weightless GCN) for MI455X / gfx1250.
//
// Roofline: no dense GEMM exists (weight=False). Work = LN/ReLU streaming
// (25.6MB/pass), L2-resident edge gathers (x fits in 192MB L2), and ~205M
// f32 scatter-add atomics -> hardware global_atomic_add_f32 is the critical
// path. WMMA is used where it genuinely applies: the LayerNorm row
// reductions, computed exactly in f32 with V_WMMA_F32_16X16X4_F32 against an
// all-ones B matrix (D = A x 1 = row sums), fed from an LDS-staged,
// bank-padded 16x128 tile in the documented A-matrix layout.
// ---------------------------------------------------------------------------

#define NN      50000
#define DF      128
#define LN_EPSF 1e-5f

typedef float v2f __attribute__((ext_vector_type(2)));
typedef float v8f __attribute__((ext_vector_type(8)));

__device__ __forceinline__ void atomAddF(float* p, float v) {
  unsafeAtomicAdd(p, v);   // hardware global_atomic_add_f32 (no CAS loop)
}

// ---- zero workspace regions ------------------------------------------------
__global__ void zero_kernel(float* __restrict__ p, long n) {
  long i = (long)blockIdx.x * blockDim.x + threadIdx.x;
  long s = (long)gridDim.x * blockDim.x;
  for (; i < n; i += s) p[i] = 0.0f;
}

// ---- degree histogram ------------------------------------------------------
__global__ void degree_kernel(const int* __restrict__ src, const int* __restrict__ dst,
                              float* __restrict__ degout, float* __restrict__ degin,
                              int ne) {
  int i = blockIdx.x * blockDim.x + threadIdx.x;
  int s = gridDim.x * blockDim.x;
  for (; i < ne; i += s) {
    atomAddF(&degout[src[i]], 1.0f);
    atomAddF(&degin [dst[i]], 1.0f);
  }
}

// ---- deg -> rsqrt(clamp(deg,1)) in place -----------------------------------
__global__ void rsqrt_kernel(float* __restrict__ d, int n) {
  int i = blockIdx.x * blockDim.x + threadIdx.x;
  if (i < n) { float v = d[i]; d[i] = rsqrtf(v < 1.0f ? 1.0f : v); }
}

// ---- fused (prescale + residual) -> LayerNorm -> ReLU -> out-scale ---------
// One wave owns a 16-node tile. Tile staged coalesced into LDS with padded
// row stride (132 floats) to spread banks; row sums and row sum-of-squares
// computed with V_WMMA_F32_16X16X4_F32 against an all-ones B (exact f32).
#define LN_WAVES   4
#define ROWSTRIDE  132                        // 128 + 4 pad floats
#define TILE_FL    (16 * ROWSTRIDE + 32)      // tile + mu[16] + rs[16]

__global__ __launch_bounds__(128) void ln_wmma_kernel(
    const float* __restrict__ in,       // node features (pre-norm input)
    const float* __restrict__ pre_rs,   // optional per-node prescale (din^-1/2), or null
    const float* __restrict__ resid,    // optional residual (original x), or null
    const float* __restrict__ out_rs,   // per-node output scale (dout^-1/2)
    float* __restrict__ h,              // output: relu(LN(.)) * out_rs
    int ntiles)
{
  __shared__ float lds[LN_WAVES * TILE_FL];
  const int lane = threadIdx.x & 31;
  const int wave = threadIdx.x >> 5;

  int  tile    = blockIdx.x * LN_WAVES + wave;
  bool tile_ok = (tile < ntiles);
  if (!tile_ok) tile = ntiles - 1;            // keep full wave active for WMMA

  float* T   = lds + wave * TILE_FL;
  float* muB = T + 16 * ROWSTRIDE;
  float* rsB = muB + 16;

  // --- stage 16x128 tile into LDS, coalesced (512B per row request) --------
  for (int r = 0; r < 16; ++r) {
    int node = tile * 16 + r; if (node >= NN) node = NN - 1;
    float s = pre_rs ? pre_rs[node] : 1.0f;
    float4 v = ((const float4*)(in + (long)node * DF))[lane];
    v.x *= s; v.y *= s; v.z *= s; v.w *= s;
    if (resid) {
      float4 vr = ((const float4*)(resid + (long)node * DF))[lane];
      v.x += vr.x; v.y += vr.y; v.z += vr.z; v.w += vr.w;
    }
    ((float4*)(T + r * ROWSTRIDE))[lane] = v;  // 16B aligned (528*r + 16*lane)
  }
  __syncthreads();

  // --- WMMA row statistics --------------------------------------------------
  // A-matrix 16x4 f32 layout: lanes 0-15 = rows 0-15 holding K0,K1;
  // lanes 16-31 = rows 0-15 holding K2,K3.  B = all ones -> D = row sums.
  const int lrow  = lane & 15;
  const int khalf = lane >> 4;
  v2f ones; ones.x = 1.0f; ones.y = 1.0f;
  v8f acc  = {};
  v8f accq = {};
  for (int kc = 0; kc < 32; ++kc) {
    v2f a = *(const v2f*)(T + lrow * ROWSTRIDE + kc * 4 + khalf * 2);
    v2f aq; aq.x = a.x * a.x; aq.y = a.y * a.y;
    acc  = __builtin_amdgcn_wmma_f32_16x16x4_f32(false, a,  false, ones,
                                                 (short)0, acc,  false, false);
    accq = __builtin_amdgcn_wmma_f32_16x16x4_f32(false, aq, false, ones,
                                                 (short)0, accq, false, false);
  }
  // C/D layout: VGPR v -> row v (lanes 0-15), row v+8 (lanes 16-31); value
  // replicated over all N columns, so lane lrow==0 of each half publishes.
  if (lrow == 0) {
    int base = khalf * 8;
    for (int v = 0; v < 8; ++v) {
      float mu  = acc[v]  * (1.0f / DF);
      float var = accq[v] * (1.0f / DF) - mu * mu;
      muB[base + v] = mu;
      rsB[base + v] = rsqrtf(var + LN_EPSF);
    }
  }
  __syncthreads();

  // --- normalize + ReLU + per-node out scale, write back into LDS ----------
  {
    int node = tile * 16 + lrow; if (node >= NN) node = NN - 1;
    float mu = muB[lrow];
    float rs = rsB[lrow];
    float sc = out_rs[node];
    for (int kc = 0; kc < 32; ++kc) {
      float* p = T + lrow * ROWSTRIDE + kc * 4 + khalf * 2;
      v2f a = *(const v2f*)p;
      float y0 = (a.x - mu) * rs; y0 = (y0 > 0.0f) ? y0 * sc : 0.0f;
      float y1 = (a.y - mu) * rs; y1 = (y1 > 0.0f) ? y1 * sc : 0.0f;
      v2f y; y.x = y0; y.y = y1;
      *(v2f*)p = y;
    }
  }
  __syncthreads();

  // --- coalesced copy-out ---------------------------------------------------
  for (int r = 0; r < 16; ++r) {
    int node = tile * 16 + r;
    if (tile_ok && node < NN) {
      float4 v = ((const float4*)(T + r * ROWSTRIDE))[lane];
      ((float4*)(h + (long)node * DF))[lane] = v;
    }
  }
}

// ---- edge scatter-add: one wave per edge, lane = float4 slice --------------
__global__ void scatter_kernel(const int* __restrict__ src, const int* __restrict__ dst,
                               const float* __restrict__ h, float* __restrict__ agg,
                               int ne) {
  const int lane = threadIdx.x & 31;
  long w  = (long)blockIdx.x * (blockDim.x >> 5) + (threadIdx.x >> 5);
  long nw = (long)gridDim.x * (blockDim.x >> 5);
  for (long e = w; e < ne; e += nw) {
    int s = src[e];
    int d = dst[e];
    float4 v = ((const float4*)(h + (long)s * DF))[lane];   // 512B coalesced, L2 hit
    float* base = agg + (long)d * DF + lane * 4;
    atomAddF(base + 0, v.x);
    atomAddF(base + 1, v.y);
    atomAddF(base + 2, v.z);
    atomAddF(base + 3, v.w);
  }
}

// ---- final per-node din^-1/2 scale of the aggregated output ----------------
__global__ void scale_kernel(float* __restrict__ out, const float* __restrict__ rs) {
  long n = (long)NN * DF;
  long i = (long)blockIdx.x * blockDim.x + threadIdx.x;
  long s = (long)gridDim.x * blockDim.x;
  for (; i < n; i += s) out[i] *= rs[i >> 7];
}

// ---------------------------------------------------------------------------
extern "C" void kernel_launch(void* const* d_in, const int* in_sizes, int n_in,
                              void* d_out, int out_size, void* d_ws, size_t ws_size,
                              hipStream_t stream) {
  const float* x  = (const float*)d_in[0];
  const int*   ei = (const int*)d_in[1];          // edge_index, shape (2, E) flat
  const int    ne = in_sizes[1] / 2;              // 800000
  const int*   src = ei;
  const int*   dst = ei + ne;

  float* ws     = (float*)d_ws;
  float* degout = ws;                             // 50000 (+16 pad) -> dout^-1/2
  float* degin  = ws + 50016;                     // 50000 (+16 pad) -> din^-1/2
  float* h      = ws + 100032;                    // 50000*128
  float* agg1   = h + (long)NN * DF;              // 50000*128
  float* out    = (float*)d_out;

  const long nfeat = (long)NN * DF;

  // 1) zero degrees (incl. pads), layer-1 accumulator, and output accumulator
  zero_kernel<<<256, 256, 0, stream>>>(degout, 100032);
  zero_kernel<<<2048, 256, 0, stream>>>(agg1, nfeat);
  zero_kernel<<<2048, 256, 0, stream>>>(out, nfeat);

  // 2) degree histogram, 3) rsqrt(clamp(deg,1)) in place
  degree_kernel<<<1024, 256, 0, stream>>>(src, dst, degout, degin, ne);
  rsqrt_kernel<<<(100032 + 255) / 256, 256, 0, stream>>>(degout, 100032);

  const int ntiles   = (NN + 15) / 16;                        // 3125
  const int lnblocks = (ntiles + LN_WAVES - 1) / LN_WAVES;    // 782

  // Layer 1: h = relu(LN(x)) * dout^-1/2 ; agg1 = scatter-sum over edges
  ln_wmma_kernel<<<lnblocks, 128, 0, stream>>>(x, nullptr, nullptr, degout, h, ntiles);
  scatter_kernel<<<2048, 256, 0, stream>>>(src, dst, h, agg1, ne);

  // Layer 2: t = agg1*din^-1/2 + x ; h = relu(LN(t)) * dout^-1/2 ; out = scatter
  ln_wmma_kernel<<<lnblocks, 128, 0, stream>>>(agg1, degin, x, degout, h, ntiles);
  scatter_kernel<<<2048, 256, 0, stream>>>(src, dst, h, out, ne);

  // Final: out *= din^-1/2
  scale_kernel<<<2048, 256, 0, stream>>>(out, degin);
}